// LightGCN_43207370998363
// MI455X (gfx1250) — compile-verified
//
#include <hip/hip_runtime.h>
#include <hip/hip_bf16.h>

#define N_USERS 100000
#define N_ITEMS 50000
#define N_NODES 150000
#define N_EDGES 4800000
#define EMB_DIM 64
#define TILE 256

static_assert(N_EDGES % TILE == 0, "exact tiling");
static_assert(N_NODES == N_USERS + N_ITEMS, "node count");

// ---- CDNA5 async global->LDS path (guarded so it compiles on both toolchains) ----
#if defined(__has_builtin)
#  if __has_builtin(__builtin_amdgcn_global_load_async_to_lds_b32)
#    define HAVE_ASYNC_LDS 1
#  endif
#  if __has_builtin(__builtin_amdgcn_s_wait_asynccnt)
#    define HAVE_WAIT_ASYNC 1
#  endif
#endif

#if defined(HAVE_ASYNC_LDS)
#if defined(HAVE_WAIT_ASYNC)
#define WAIT_ASYNC(n) __builtin_amdgcn_s_wait_asynccnt(n)
#else
#define WAIT_ASYNC(n) asm volatile("s_wait_asynccnt %0" :: "i"(n) : "memory")
#endif

__device__ __forceinline__ void stage_async(const int* esrc, const int* edst,
                                            const float* ew, int base, int t,
                                            int* ls, int* ld, float* lw) {
    __builtin_amdgcn_global_load_async_to_lds_b32(
        const_cast<int*>(esrc) + base + t, ls + t, 0, 0);
    __builtin_amdgcn_global_load_async_to_lds_b32(
        const_cast<int*>(edst) + base + t, ld + t, 0, 0);
    __builtin_amdgcn_global_load_async_to_lds_b32(
        reinterpret_cast<int*>(const_cast<float*>(ew)) + base + t,
        reinterpret_cast<int*>(lw) + t, 0, 0);
}
#endif

// ---------------------------------------------------------------------------
// SpMM: xout[dst] += w * xin[src]   (xout pre-zeroed)
// 16 lanes per edge, float4 per lane; edge stream staged into LDS
// asynchronously, double-buffered. Placed first so --disasm snippet shows it.
// ---------------------------------------------------------------------------
__global__ __launch_bounds__(256)
void lgcn_spmm(const float* __restrict__ xin, float* __restrict__ xout,
               const int* __restrict__ esrc, const int* __restrict__ edst,
               const float* __restrict__ ew) {
    __shared__ int   s_src[2][TILE];
    __shared__ int   s_dst[2][TILE];
    __shared__ float s_w[2][TILE];

    const int NT = N_EDGES / TILE;
    const int t = threadIdx.x;
    const int stride = gridDim.x;
    int tile = blockIdx.x;
    if (tile >= NT) return;

#if defined(HAVE_ASYNC_LDS)
    stage_async(esrc, edst, ew, tile * TILE, t, s_src[0], s_dst[0], s_w[0]);
#endif
    int buf = 0;
    const float4* x4 = (const float4*)xin;
    const int g = t >> 4;   // 16 edge-groups of 16 lanes per block
    const int j = t & 15;   // lane within group: 4 floats each

    while (tile < NT) {
        const int nxt = tile + stride;
#if defined(HAVE_ASYNC_LDS)
        if (nxt < NT) {
            stage_async(esrc, edst, ew, nxt * TILE, t,
                        s_src[buf ^ 1], s_dst[buf ^ 1], s_w[buf ^ 1]);
            WAIT_ASYNC(3);   // async loads complete in order: current tile's 3 done
        } else {
            WAIT_ASYNC(0);
        }
        __syncthreads();
#else
        s_src[buf][t] = esrc[tile * TILE + t];
        s_dst[buf][t] = edst[tile * TILE + t];
        s_w[buf][t]   = ew[tile * TILE + t];
        __syncthreads();
#endif
        #pragma unroll 4
        for (int k = 0; k < 16; ++k) {
            const int e = (g << 4) | k;
            const int s  = s_src[buf][e];
            const int d  = s_dst[buf][e];
            const float wt = s_w[buf][e];
            float4 v = x4[s * (EMB_DIM / 4) + j];     // global_load_b128 gather
            float* o = xout + d * EMB_DIM + (j << 2);
            __hip_atomic_fetch_add(o + 0, wt * v.x, __ATOMIC_RELAXED, __HIP_MEMORY_SCOPE_AGENT);
            __hip_atomic_fetch_add(o + 1, wt * v.y, __ATOMIC_RELAXED, __HIP_MEMORY_SCOPE_AGENT);
            __hip_atomic_fetch_add(o + 2, wt * v.z, __ATOMIC_RELAXED, __HIP_MEMORY_SCOPE_AGENT);
            __hip_atomic_fetch_add(o + 3, wt * v.w, __ATOMIC_RELAXED, __HIP_MEMORY_SCOPE_AGENT);
        }
        __syncthreads();   // all waves done reading buf before it is re-staged
        tile = nxt;
        buf ^= 1;
    }
}

// ---------------------------------------------------------------------------
// init: x0 = acc = concat(user_emb, item_emb); x1 = 0
// ---------------------------------------------------------------------------
__global__ __launch_bounds__(256)
void lgcn_init(const float* __restrict__ ue, const float* __restrict__ ie,
               float* __restrict__ x0, float* __restrict__ x1,
               float* __restrict__ acc) {
    const int total4 = N_NODES * EMB_DIM / 4;
    int i = blockIdx.x * blockDim.x + threadIdx.x;
    if (i >= total4) return;
    const float4* u4 = (const float4*)ue;
    const float4* i4 = (const float4*)ie;
    const int ucut = N_USERS * (EMB_DIM / 4);
    float4 v = (i < ucut) ? u4[i] : i4[i - ucut];
    ((float4*)x0)[i] = v;
    ((float4*)acc)[i] = v;
    float4 z = {0.f, 0.f, 0.f, 0.f};
    ((float4*)x1)[i] = z;
}

// ---------------------------------------------------------------------------
// accum: acc += xr; xz = 0  (prepares next layer's output buffer)
// ---------------------------------------------------------------------------
__global__ __launch_bounds__(256)
void lgcn_accum_zero(float* __restrict__ acc, const float* __restrict__ xr,
                     float* __restrict__ xz) {
    const int total4 = N_NODES * EMB_DIM / 4;
    int i = blockIdx.x * blockDim.x + threadIdx.x;
    if (i >= total4) return;
    float4 a = ((float4*)acc)[i];
    float4 r = ((const float4*)xr)[i];
    a.x += r.x; a.y += r.y; a.z += r.z; a.w += r.w;
    ((float4*)acc)[i] = a;
    float4 z = {0.f, 0.f, 0.f, 0.f};
    ((float4*)xz)[i] = z;
}

// ---------------------------------------------------------------------------
// final: out = (acc + last_layer) / 4
// ---------------------------------------------------------------------------
__global__ __launch_bounds__(256)
void lgcn_final(float* __restrict__ acc, const float* __restrict__ xr) {
    const int total4 = N_NODES * EMB_DIM / 4;
    int i = blockIdx.x * blockDim.x + threadIdx.x;
    if (i >= total4) return;
    float4 a = ((float4*)acc)[i];
    float4 r = ((const float4*)xr)[i];
    a.x = (a.x + r.x) * 0.25f;
    a.y = (a.y + r.y) * 0.25f;
    a.z = (a.z + r.z) * 0.25f;
    a.w = (a.w + r.w) * 0.25f;
    ((float4*)acc)[i] = a;
}

extern "C" void kernel_launch(void* const* d_in, const int* in_sizes, int n_in,
                              void* d_out, int out_size, void* d_ws, size_t ws_size,
                              hipStream_t stream) {
    const float* user_emb = (const float*)d_in[0];
    const float* item_emb = (const float*)d_in[1];
    const float* ew       = (const float*)d_in[2];
    const int*   esrc     = (const int*)d_in[3];
    const int*   edst     = (const int*)d_in[4];
    float* out = (float*)d_out;                 // acc lives here (150000 x 64)

    float* x0 = (float*)d_ws;                   // 38.4 MB
    float* x1 = x0 + (size_t)N_NODES * EMB_DIM; // 38.4 MB

    const int total4 = N_NODES * EMB_DIM / 4;
    const int eb = 256;
    const int eg = (total4 + eb - 1) / eb;

    const int spmm_blocks = 2048;               // ~9 tiles per block -> async pipeline pays off

    lgcn_init<<<eg, eb, 0, stream>>>(user_emb, item_emb, x0, x1, out);

    // layer 1: x0 -> x1
    lgcn_spmm<<<spmm_blocks, 256, 0, stream>>>(x0, x1, esrc, edst, ew);
    lgcn_accum_zero<<<eg, eb, 0, stream>>>(out, x1, x0);   // acc += l1; zero x0
    // layer 2: x1 -> x0
    lgcn_spmm<<<spmm_blocks, 256, 0, stream>>>(x1, x0, esrc, edst, ew);
    lgcn_accum_zero<<<eg, eb, 0, stream>>>(out, x0, x1);   // acc += l2; zero x1
    // layer 3: x0 -> x1
    lgcn_spmm<<<spmm_blocks, 256, 0, stream>>>(x0, x1, esrc, edst, ew);
    lgcn_final<<<eg, eb, 0, stream>>>(out, x1);            // out = (acc + l3)/4
}